// ELAB_53180285059592
// MI455X (gfx1250) — compile-verified
//
#include <hip/hip_runtime.h>
#include <math.h>

typedef __attribute__((ext_vector_type(2))) float v2f;
typedef __attribute__((ext_vector_type(8))) float v8f;

// Problem constants (match reference)
#define BB   4
#define CH   60
#define CE   120
#define HH   256
#define WW   256
#define WSZ  8
#define NPIX (BB * HH * WW)          // 262144
#define NH   (HH / WSZ)              // 32
#define NW   (WW / WSZ)              // 32

// ---------------------------------------------------------------------------
// Fused 1x1 conv (channel GEMM) with optional input shift, GELU, BN, residual.
// One wave computes all M-tiles of a 16-pixel column; weights staged in LDS.
//   out[o, p] = act( bn( sum_c W[o,c] * shift(in)[c, p] + bias[o] ) ) + resid
// WMMA: V_WMMA_F32_16X16X4_F32, M=out-chan tile, N=16 pixels, K=4 in-chans.
// ---------------------------------------------------------------------------
template <int CIN, int COUT, bool SHIFT, bool GELU_ACT, bool BN, bool RESID>
__global__ __launch_bounds__(256) void conv1x1_k(
    const float* __restrict__ in, const float* __restrict__ wgt,
    const float* __restrict__ bias, const float* __restrict__ resid,
    const float* __restrict__ bng, const float* __restrict__ bnb,
    const float* __restrict__ bnm, const float* __restrict__ bnv,
    float* __restrict__ out) {
  constexpr int MT = (COUT + 15) / 16;  // M tiles (rows padded with zeros)
  constexpr int KS = CIN / 4;           // K steps (CIN divisible by 4)

  __shared__ float sw[MT * 16][CIN + 1];  // zero-padded weight rows
  __shared__ float smul[MT * 16];         // fused epilogue scale
  __shared__ float sadd[MT * 16];         // fused epilogue offset

  const int tid = threadIdx.x;

  for (int i = tid; i < MT * 16 * CIN; i += 256) {
    int m = i / CIN, c = i % CIN;
    sw[m][c] = (m < COUT) ? wgt[m * CIN + c] : 0.0f;
  }
  for (int i = tid; i < MT * 16; i += 256) {
    float mul = 1.0f, add = 0.0f;
    if (i < COUT) {
      float bs = bias[i];
      if constexpr (BN) {
        float s = bng[i] * rsqrtf(bnv[i] + 1e-5f);
        mul = s;
        add = (bs - bnm[i]) * s + bnb[i];
      } else {
        add = bs;
      }
    }
    smul[i] = mul;
    sadd[i] = add;
  }
  __syncthreads();

  const int wave = tid >> 5;
  const int lane = tid & 31;
  const int lm = lane & 15;
  const int hi = lane >> 4;  // 0: K lanes {0,1}; 1: K lanes {2,3}

  const long p0 = ((long)blockIdx.x * 8 + wave) * 16;
  const long p = p0 + lm;
  const int wcoord = (int)(p & (WW - 1));
  const int hcoord = (int)((p >> 8) & (HH - 1));
  const int bcoord = (int)(p >> 16);

  auto loadIn = [&](int c) -> float {
    int hh = hcoord, ww = wcoord;
    if constexpr (SHIFT) {
      constexpr int g = CIN / 5;
      int grp = c / g;
      if (grp == 0)      ww += 1;
      else if (grp == 1) ww -= 1;
      else if (grp == 2) hh += 1;
      else if (grp == 3) hh -= 1;
    }
    if ((unsigned)hh >= (unsigned)HH || (unsigned)ww >= (unsigned)WW) return 0.0f;
    return in[(((size_t)bcoord * CIN + c) * HH + hh) * WW + ww];
  };

  v8f acc[MT];
#pragma unroll
  for (int mt = 0; mt < MT; ++mt)
#pragma unroll
    for (int r = 0; r < 8; ++r) acc[mt][r] = 0.0f;

  for (int ks = 0; ks < KS; ++ks) {
    const int cb = ks * 4;
    v2f bf;
    bf.x = loadIn(cb + 2 * hi);
    bf.y = loadIn(cb + 2 * hi + 1);
#pragma unroll
    for (int mt = 0; mt < MT; ++mt) {
      v2f af;
      af.x = sw[mt * 16 + lm][cb + 2 * hi];
      af.y = sw[mt * 16 + lm][cb + 2 * hi + 1];
      acc[mt] = __builtin_amdgcn_wmma_f32_16x16x4_f32(
          false, af, false, bf, (short)0, acc[mt], false, false);
    }
  }

#pragma unroll
  for (int mt = 0; mt < MT; ++mt) {
#pragma unroll
    for (int r = 0; r < 8; ++r) {
      int m = mt * 16 + r + hi * 8;
      if (m >= COUT) continue;
      float v = acc[mt][r] * smul[m] + sadd[m];
      if constexpr (GELU_ACT) v = 0.5f * v * (1.0f + erff(v * 0.70710678118654752f));
      size_t idx = (((size_t)bcoord * COUT + m) * HH + hcoord) * WW + wcoord;
      if constexpr (RESID) v += resid[idx];
      out[idx] = v;
    }
  }
}

// ---------------------------------------------------------------------------
// Window self-attention core: one workgroup (8 waves) per 8x8 window.
//   Q[o,n] = mw . sub      (60x60 x 60x64, strided window gather)
//   S      = softmax(Q^T Q)  (64x64)
//   out    = (S V)^T         written back as [c, h, w]
// All three GEMMs via V_WMMA_F32_16X16X4_F32; Q and S live in LDS.
// ---------------------------------------------------------------------------
__global__ __launch_bounds__(256) void wsa_attn_k(
    const float* __restrict__ t, const float* __restrict__ mw,
    const float* __restrict__ mb, float* __restrict__ out) {
  __shared__ float smw[64][64];   // mask weight, zero padded to 64x64
  __shared__ float smb[64];
  __shared__ float sQ[64][65];    // Q stored [o][n]
  __shared__ float sS[64][65];    // S stored [n][n2]

  const int tid = threadIdx.x;
  const int wave = tid >> 5;
  const int lane = tid & 31;
  const int lm = lane & 15;
  const int hi = lane >> 4;

  const int widx = blockIdx.x;
  const int iw = widx & (NW - 1);
  const int ih = (widx / NW) & (NH - 1);
  const int bb = widx / (NH * NW);

  for (int i = tid; i < 64 * 64; i += 256) {
    int o = i >> 6, c = i & 63;
    smw[o][c] = (o < CH && c < CH) ? mw[o * CH + c] : 0.0f;
  }
  for (int i = tid; i < 64; i += 256) smb[i] = (i < CH) ? mb[i] : 0.0f;
  __syncthreads();

  // strided/mirror-padded query sample: n=(p,q) -> xp[ih*8+2q, iw*8+2p]
  auto loadSub = [&](int c, int n) -> float {
    int pp = n >> 3, qq = n & 7;
    int row = ih * WSZ + 2 * qq;
    int col = iw * WSZ + 2 * pp;
    if (row >= HH) row -= WSZ;
    if (col >= WW) col -= WSZ;
    return t[(((size_t)bb * CH + c) * HH + row) * WW + col];
  };
  // dense value sample: n2 -> window (row = n2/8, col = n2%8)
  auto loadV = [&](int c, int n2) -> float {
    if (c >= CH) return 0.0f;
    int row = ih * WSZ + (n2 >> 3);
    int col = iw * WSZ + (n2 & 7);
    return t[(((size_t)bb * CH + c) * HH + row) * WW + col];
  };

  // ---- Phase 1: Q[o][n], 4x4 tiles, K = 60 -------------------------------
  for (int tile = wave * 2; tile < wave * 2 + 2; ++tile) {
    const int mt = tile >> 2, nt = tile & 3;
    v8f acc;
#pragma unroll
    for (int r = 0; r < 8; ++r) acc[r] = 0.0f;
    for (int ks = 0; ks < CH / 4; ++ks) {
      const int cb = ks * 4;
      v2f af, bf;
      af.x = smw[mt * 16 + lm][cb + 2 * hi];
      af.y = smw[mt * 16 + lm][cb + 2 * hi + 1];
      bf.x = loadSub(cb + 2 * hi, nt * 16 + lm);
      bf.y = loadSub(cb + 2 * hi + 1, nt * 16 + lm);
      acc = __builtin_amdgcn_wmma_f32_16x16x4_f32(
          false, af, false, bf, (short)0, acc, false, false);
    }
#pragma unroll
    for (int r = 0; r < 8; ++r) {
      int m = mt * 16 + r + hi * 8;
      sQ[m][nt * 16 + lm] = acc[r] + smb[m];  // padded rows: 0 + 0
    }
  }
  __syncthreads();

  // ---- Phase 2: S[n][n2] = sum_o Q[o][n] * Q[o][n2], K = 64 --------------
  for (int tile = wave * 2; tile < wave * 2 + 2; ++tile) {
    const int mt = tile >> 2, nt = tile & 3;
    v8f acc;
#pragma unroll
    for (int r = 0; r < 8; ++r) acc[r] = 0.0f;
    for (int ks = 0; ks < 16; ++ks) {
      const int kb = ks * 4;
      v2f af, bf;
      af.x = sQ[kb + 2 * hi][mt * 16 + lm];
      af.y = sQ[kb + 2 * hi + 1][mt * 16 + lm];
      bf.x = sQ[kb + 2 * hi][nt * 16 + lm];
      bf.y = sQ[kb + 2 * hi + 1][nt * 16 + lm];
      acc = __builtin_amdgcn_wmma_f32_16x16x4_f32(
          false, af, false, bf, (short)0, acc, false, false);
    }
#pragma unroll
    for (int r = 0; r < 8; ++r)
      sS[mt * 16 + r + hi * 8][nt * 16 + lm] = acc[r];
  }
  __syncthreads();

  // ---- Phase 3: row softmax over sS --------------------------------------
  if (tid < 64) {
    float mx = -3.4e38f;
    for (int j = 0; j < 64; ++j) mx = fmaxf(mx, sS[tid][j]);
    float sum = 0.0f;
    for (int j = 0; j < 64; ++j) {
      float e = __expf(sS[tid][j] - mx);
      sS[tid][j] = e;
      sum += e;
    }
    float inv = 1.0f / sum;
    for (int j = 0; j < 64; ++j) sS[tid][j] *= inv;
  }
  __syncthreads();

  // ---- Phase 4: out[c][n] = sum_n2 V[c][n2] * S[n][n2], K = 64 -----------
  for (int tile = wave * 2; tile < wave * 2 + 2; ++tile) {
    const int mt = tile >> 2, nt = tile & 3;  // mt: channel tile, nt: position tile
    v8f acc;
#pragma unroll
    for (int r = 0; r < 8; ++r) acc[r] = 0.0f;
    for (int ks = 0; ks < 16; ++ks) {
      const int kb = ks * 4;
      v2f af, bf;
      af.x = loadV(mt * 16 + lm, kb + 2 * hi);
      af.y = loadV(mt * 16 + lm, kb + 2 * hi + 1);
      bf.x = sS[nt * 16 + lm][kb + 2 * hi];        // B = S^T
      bf.y = sS[nt * 16 + lm][kb + 2 * hi + 1];
      acc = __builtin_amdgcn_wmma_f32_16x16x4_f32(
          false, af, false, bf, (short)0, acc, false, false);
    }
#pragma unroll
    for (int r = 0; r < 8; ++r) {
      int c = mt * 16 + r + hi * 8;
      if (c >= CH) continue;
      int n = nt * 16 + lm;
      int row = ih * WSZ + (n >> 3);
      int col = iw * WSZ + (n & 7);
      out[(((size_t)bb * CH + c) * HH + row) * WW + col] = acc[r];
    }
  }
}

// ---------------------------------------------------------------------------
extern "C" void kernel_launch(void* const* d_in, const int* in_sizes, int n_in,
                              void* d_out, int out_size, void* d_ws,
                              size_t ws_size, hipStream_t stream) {
  (void)in_sizes; (void)n_in; (void)out_size; (void)ws_size;
  const float* x      = (const float*)d_in[0];
  const float* lfe_w0 = (const float*)d_in[1];
  const float* lfe_b0 = (const float*)d_in[2];
  const float* lfe_w1 = (const float*)d_in[3];
  const float* lfe_b1 = (const float*)d_in[4];
  const float* pi_w   = (const float*)d_in[5];
  const float* pi_b   = (const float*)d_in[6];
  const float* bn_g   = (const float*)d_in[7];
  const float* bn_b   = (const float*)d_in[8];
  const float* bn_m   = (const float*)d_in[9];
  const float* bn_v   = (const float*)d_in[10];
  const float* mask_w = (const float*)d_in[11];
  const float* mask_b = (const float*)d_in[12];
  const float* po_w   = (const float*)d_in[13];
  const float* po_b   = (const float*)d_in[14];
  float* out = (float*)d_out;

  float* ws = (float*)d_ws;
  float* bufA    = ws;                                 // x2   : CH*NPIX
  float* bufT    = ws + (size_t)CH * NPIX;             // t    : CH*NPIX
  float* bufAttn = bufT + (size_t)CH * NPIX;           // attn : CH*NPIX
  float* bufY    = bufT;                               // y1   : CE*NPIX (spans T+Attn)

  const int CONV_GRID = NPIX / (16 * 8);  // 2048 blocks, 8 waves x 16 pixels
  const int ATTN_GRID = BB * NH * NW;     // 4096 windows

  for (int i = 0; i < 2; ++i) {
    const float* Xin = (i == 0) ? x : out;

    // 1) y1 = gelu(conv1x1(shift(x), w0, b0))            C -> Ce
    conv1x1_k<CH, CE, true, true, false, false><<<CONV_GRID, 256, 0, stream>>>(
        Xin, lfe_w0 + (size_t)i * CE * CH, lfe_b0 + (size_t)i * CE, nullptr,
        nullptr, nullptr, nullptr, nullptr, bufY);

    // 2) x2 = conv1x1(shift(y1), w1, b1) + x             Ce -> C (+resid)
    conv1x1_k<CE, CH, true, false, false, true><<<CONV_GRID, 256, 0, stream>>>(
        bufY, lfe_w1 + (size_t)i * CH * CE, lfe_b1 + (size_t)i * CH, Xin,
        nullptr, nullptr, nullptr, nullptr, bufA);

    // 3) t = BN(conv1x1(x2, pi_w, pi_b))                 C -> C (+BN)
    conv1x1_k<CH, CH, false, false, true, false><<<CONV_GRID, 256, 0, stream>>>(
        bufA, pi_w + (size_t)i * CH * CH, pi_b + (size_t)i * CH, nullptr,
        bn_g + (size_t)i * CH, bn_b + (size_t)i * CH, bn_m + (size_t)i * CH,
        bn_v + (size_t)i * CH, bufT);

    // 4) a = windowed self-attention(t)
    wsa_attn_k<<<ATTN_GRID, 256, 0, stream>>>(
        bufT, mask_w + (size_t)i * CH * CH, mask_b + (size_t)i * CH, bufAttn);

    // 5) x = conv1x1(a, po_w, po_b) + x2                 C -> C (+resid)
    conv1x1_k<CH, CH, false, false, false, true><<<CONV_GRID, 256, 0, stream>>>(
        bufAttn, po_w + (size_t)i * CH * CH, po_b + (size_t)i * CH, bufA,
        nullptr, nullptr, nullptr, nullptr, out);
  }
}